// SensorMeshToFlowFrontModel_5746666242098
// MI455X (gfx1250) — compile-verified
//
#include <hip/hip_runtime.h>
#include <hip/hip_bf16.h>

#define NNODES 524288
#define NEDGES 1572864

typedef __attribute__((ext_vector_type(2))) float v2f;
typedef __attribute__((ext_vector_type(8))) float v8f;

// ---------------------------------------------------------------------------
// Fused dual-GEMM per layer:  Y = act(in) @ W0^T + b0 ; NBR = act(in) @ W1^T + b1
// One wave = one 16-node tile; V_WMMA_F32_16X16X4_F32, K stepped by 4.
// Y is written straight into the aggregation buffer (agg starts at Y).
// ---------------------------------------------------------------------------
template<int DIN, int DOUT, bool RELU>
__global__ __launch_bounds__(128) void gemm2_wmma_kernel(
    const float* __restrict__ in,
    const float* __restrict__ W0p, const float* __restrict__ b0p,
    const float* __restrict__ W1p, const float* __restrict__ b1p,
    float* __restrict__ outY, float* __restrict__ outN)
{
    constexpr int NT = (DOUT + 15) / 16;   // 16-wide N tiles
    constexpr int KC = (DIN + 3) / 4;      // K chunks of 4

    const int wave     = blockIdx.x * (blockDim.x >> 5) + (threadIdx.x >> 5);
    const int lane     = threadIdx.x & 31;
    const int nodeBase = wave * 16;        // N divisible by 16: always full tiles
    const int m        = lane & 15;        // row (A/D) or column (B/D) index
    const int kh       = (lane >> 4) << 1; // K sub-offset: 0 (lanes 0-15) / 2 (16-31)
    const int mlo      = (lane >> 4) << 3; // D rows: r -> m = r + mlo

    v8f accY[NT];
    v8f accN[NT];
#pragma unroll
    for (int t = 0; t < NT; ++t) {
        const int n = t * 16 + m;
        const float by = (n < DOUT) ? b0p[n] : 0.0f;
        const float bn = (n < DOUT) ? b1p[n] : 0.0f;
#pragma unroll
        for (int r = 0; r < 8; ++r) { accY[t][r] = by; accN[t][r] = bn; }
    }

    const float* __restrict__ row = in + (size_t)(nodeBase + m) * DIN;

#pragma unroll
    for (int kc = 0; kc < KC; ++kc) {
        const int k = kc * 4;
        v2f a; a.x = 0.0f; a.y = 0.0f;
        if (k + kh     < DIN) a.x = row[k + kh];
        if (k + kh + 1 < DIN) a.y = row[k + kh + 1];
        if (RELU) { a.x = fmaxf(a.x, 0.0f); a.y = fmaxf(a.y, 0.0f); }

#pragma unroll
        for (int t = 0; t < NT; ++t) {
            const int n = t * 16 + m;
            v2f bw0; bw0.x = 0.0f; bw0.y = 0.0f;
            v2f bw1; bw1.x = 0.0f; bw1.y = 0.0f;
            if (n < DOUT) {
                if (k + kh < DIN) {
                    bw0.x = W0p[n * DIN + k + kh];
                    bw1.x = W1p[n * DIN + k + kh];
                }
                if (k + kh + 1 < DIN) {
                    bw0.y = W0p[n * DIN + k + kh + 1];
                    bw1.y = W1p[n * DIN + k + kh + 1];
                }
            }
            accY[t] = __builtin_amdgcn_wmma_f32_16x16x4_f32(
                false, a, false, bw0, (short)0, accY[t], false, false);
            accN[t] = __builtin_amdgcn_wmma_f32_16x16x4_f32(
                false, a, false, bw1, (short)0, accN[t], false, false);
        }
    }

    // Store D tiles: VGPR r holds row (mlo + r), column n = t*16 + (lane&15)
#pragma unroll
    for (int t = 0; t < NT; ++t) {
        const int n = t * 16 + m;
        if (n < DOUT) {
#pragma unroll
            for (int r = 0; r < 8; ++r) {
                const int mm = mlo + r;
                outY[(size_t)(nodeBase + mm) * DOUT + n] = accY[t][r];
                outN[(size_t)(nodeBase + mm) * DOUT + n] = accN[t][r];
            }
        }
    }
}

// ---------------------------------------------------------------------------
// Symmetric edge scatter: agg[src] += nbr[dst]; agg[dst] += nbr[src]
// One thread per (edge, feature); feature lanes coalesce, atomics hit L2
// (widest feature array = 134 MB < 192 MB L2). Edge pair loaded as one b64.
// ---------------------------------------------------------------------------
template<int LOGD>
__global__ __launch_bounds__(256) void scatter_add_kernel(
    const int2* __restrict__ edges, const float* __restrict__ nbr,
    float* __restrict__ agg, long long total)
{
    long long i = (long long)blockIdx.x * blockDim.x + threadIdx.x;
    if (i >= total) return;
    const int e = (int)(i >> LOGD);
    const int f = (int)(i & ((1LL << LOGD) - 1));
    const int2 ed = edges[e];                 // single global_load_b64
    const int s = ed.x;
    const int d = ed.y;
    atomicAdd(&agg[((size_t)s << LOGD) + f], nbr[((size_t)d << LOGD) + f]);
    atomicAdd(&agg[((size_t)d << LOGD) + f], nbr[((size_t)s << LOGD) + f]);
}

// ---------------------------------------------------------------------------
// Last layer: 32 -> 1 (dot products; WMMA pointless at dout=1).
// Row read as 8 x float4 -> global_load_b128.
// ---------------------------------------------------------------------------
__global__ __launch_bounds__(256) void last_layer_kernel(
    const float* __restrict__ in,
    const float* __restrict__ W0p, const float* __restrict__ b0p,
    const float* __restrict__ W1p, const float* __restrict__ b1p,
    float* __restrict__ outY, float* __restrict__ outN)
{
    int i = blockIdx.x * blockDim.x + threadIdx.x;
    if (i >= NNODES) return;
    const float4* r4 = (const float4*)(in + (size_t)i * 32);
    float y = b0p[0];
    float v = b1p[0];
#pragma unroll
    for (int q = 0; q < 8; ++q) {
        const float4 xv4 = r4[q];
        const float xs[4] = { fmaxf(xv4.x, 0.0f), fmaxf(xv4.y, 0.0f),
                              fmaxf(xv4.z, 0.0f), fmaxf(xv4.w, 0.0f) };
#pragma unroll
        for (int j = 0; j < 4; ++j) {
            const int k = q * 4 + j;
            y = fmaf(xs[j], W0p[k], y);
            v = fmaf(xs[j], W1p[k], v);
        }
    }
    outY[i] = y;
    outN[i] = v;
}

// Vectorized output pass: sigmoid over N floats via float4 (b128).
__global__ __launch_bounds__(256) void sigmoid_kernel(
    const float4* __restrict__ agg, float4* __restrict__ out)
{
    int i = blockIdx.x * blockDim.x + threadIdx.x;
    if (i >= NNODES / 4) return;
    float4 a = agg[i];
    a.x = 1.0f / (1.0f + __expf(-a.x));
    a.y = 1.0f / (1.0f + __expf(-a.y));
    a.z = 1.0f / (1.0f + __expf(-a.z));
    a.w = 1.0f / (1.0f + __expf(-a.w));
    out[i] = a;
}

// ---------------------------------------------------------------------------
extern "C" void kernel_launch(void* const* d_in, const int* in_sizes, int n_in,
                              void* d_out, int out_size, void* d_ws, size_t ws_size,
                              hipStream_t stream)
{
    const float* x     = (const float*)d_in[0];
    const int2*  edges = (const int2*) d_in[1];
    // Per-layer params: W0_i, b0_i, W1_i, b1_i at d_in[2 + 4*(i-1) ...]
    const float* W0[5]; const float* b0[5]; const float* W1[5]; const float* b1[5];
    for (int i = 0; i < 5; ++i) {
        W0[i] = (const float*)d_in[2 + 4 * i + 0];
        b0[i] = (const float*)d_in[2 + 4 * i + 1];
        W1[i] = (const float*)d_in[2 + 4 * i + 2];
        b1[i] = (const float*)d_in[2 + 4 * i + 3];
    }

    float* ws = (float*)d_ws;
    float* A = ws;                            // width up to 64
    float* B = ws + (size_t)NNODES * 64;      // width up to 64 (always NBR)
    float* C = ws + (size_t)NNODES * 128;     // width up to 32
    float* out = (float*)d_out;

    const int GEMM_BLOCKS = NNODES / 64;      // 4 waves/block * 16 nodes/wave

    // Layer 1: x(1) -> agg=A(16), nbr=B(16)
    gemm2_wmma_kernel<1, 16, false><<<GEMM_BLOCKS, 128, 0, stream>>>(
        x, W0[0], b0[0], W1[0], b1[0], A, B);
    {
        long long total = (long long)NEDGES << 4;
        scatter_add_kernel<4><<<(int)((total + 255) / 256), 256, 0, stream>>>(edges, B, A, total);
    }

    // Layer 2: relu(A)(16) -> agg=C(32), nbr=B(32)
    gemm2_wmma_kernel<16, 32, true><<<GEMM_BLOCKS, 128, 0, stream>>>(
        A, W0[1], b0[1], W1[1], b1[1], C, B);
    {
        long long total = (long long)NEDGES << 5;
        scatter_add_kernel<5><<<(int)((total + 255) / 256), 256, 0, stream>>>(edges, B, C, total);
    }

    // Layer 3: relu(C)(32) -> agg=A(64), nbr=B(64)
    gemm2_wmma_kernel<32, 64, true><<<GEMM_BLOCKS, 128, 0, stream>>>(
        C, W0[2], b0[2], W1[2], b1[2], A, B);
    {
        long long total = (long long)NEDGES << 6;
        scatter_add_kernel<6><<<(int)((total + 255) / 256), 256, 0, stream>>>(edges, B, A, total);
    }

    // Layer 4: relu(A)(64) -> agg=C(32), nbr=B(32)
    gemm2_wmma_kernel<64, 32, true><<<GEMM_BLOCKS, 128, 0, stream>>>(
        A, W0[3], b0[3], W1[3], b1[3], C, B);
    {
        long long total = (long long)NEDGES << 5;
        scatter_add_kernel<5><<<(int)((total + 255) / 256), 256, 0, stream>>>(edges, B, C, total);
    }

    // Layer 5: relu(C)(32) -> agg=A(1), nbr=B(1)  (scalar kernel)
    last_layer_kernel<<<NNODES / 256, 256, 0, stream>>>(
        C, W0[4], b0[4], W1[4], b1[4], A, B);
    {
        long long total = (long long)NEDGES;
        scatter_add_kernel<0><<<(int)((total + 255) / 256), 256, 0, stream>>>(edges, B, A, total);
    }

    // Output: sigmoid(A) -> d_out (N f32, viewed as (16, 32768))
    sigmoid_kernel<<<NNODES / 1024, 256, 0, stream>>>((const float4*)A, (float4*)out);
}